// GraphRandomWalkNPR_2989297238713
// MI455X (gfx1250) — compile-verified
//
#include <hip/hip_runtime.h>
#include <hip/hip_bf16.h>

typedef float v2f __attribute__((ext_vector_type(2)));
typedef float v8f __attribute__((ext_vector_type(8)));

#define B_ 8
#define C_ 64
#define H_ 48
#define W_ 48
#define HD_ 24
#define WD_ 24
#define N_ (H_ * W_)   /* 2304 */
#define D_ (2 * C_)    /* 128 */
#define THRESH_ 0.5f

// ---------------- Kernel 1: bilinear downsample 48->24 (align_corners=True) ----
__global__ void k_down(const float* __restrict__ x, float* __restrict__ down) {
  int idx = blockIdx.x * blockDim.x + threadIdx.x;
  const int total = B_ * C_ * HD_ * WD_;
  if (idx >= total) return;
  int ow = idx % WD_;
  int oh = (idx / WD_) % HD_;
  int bc = idx / (WD_ * HD_);
  double sh = (double)oh * (double)(H_ - 1) / (double)(HD_ - 1);
  double sw = (double)ow * (double)(W_ - 1) / (double)(WD_ - 1);
  int h0 = (int)floor(sh); if (h0 < 0) h0 = 0; if (h0 > H_ - 1) h0 = H_ - 1;
  int w0 = (int)floor(sw); if (w0 < 0) w0 = 0; if (w0 > W_ - 1) w0 = W_ - 1;
  int h1 = min(h0 + 1, H_ - 1);
  int w1 = min(w0 + 1, W_ - 1);
  float wh = (float)(sh - (double)h0);
  float ww = (float)(sw - (double)w0);
  const float* xp = x + (size_t)bc * H_ * W_;
  float v00 = xp[h0 * W_ + w0], v01 = xp[h0 * W_ + w1];
  float v10 = xp[h1 * W_ + w0], v11 = xp[h1 * W_ + w1];
  float top = v00 * (1.f - ww) + v01 * ww;
  float bot = v10 * (1.f - ww) + v11 * ww;
  down[idx] = top * (1.f - wh) + bot * wh;
}

// ---------------- Kernel 2: upsample 24->48, npr = x - blurred, build node ------
__global__ void k_npr_node(const float* __restrict__ x, const float* __restrict__ down,
                           float* __restrict__ npr_out, float* __restrict__ node) {
  int idx = blockIdx.x * blockDim.x + threadIdx.x;
  const int total = B_ * C_ * H_ * W_;
  if (idx >= total) return;
  int w = idx % W_;
  int h = (idx / W_) % H_;
  int c = (idx / (W_ * H_)) % C_;
  int b = idx / (W_ * H_ * C_);
  double sh = (double)h * (double)(HD_ - 1) / (double)(H_ - 1);
  double sw = (double)w * (double)(WD_ - 1) / (double)(W_ - 1);
  int h0 = (int)floor(sh); if (h0 < 0) h0 = 0; if (h0 > HD_ - 1) h0 = HD_ - 1;
  int w0 = (int)floor(sw); if (w0 < 0) w0 = 0; if (w0 > WD_ - 1) w0 = WD_ - 1;
  int h1 = min(h0 + 1, HD_ - 1);
  int w1 = min(w0 + 1, WD_ - 1);
  float wh = (float)(sh - (double)h0);
  float ww = (float)(sw - (double)w0);
  const float* dp = down + (size_t)(b * C_ + c) * HD_ * WD_;
  float v00 = dp[h0 * WD_ + w0], v01 = dp[h0 * WD_ + w1];
  float v10 = dp[h1 * WD_ + w0], v11 = dp[h1 * WD_ + w1];
  float top = v00 * (1.f - ww) + v01 * ww;
  float bot = v10 * (1.f - ww) + v11 * ww;
  float blurred = top * (1.f - wh) + bot * wh;
  float xv = x[idx];
  float nv = xv - blurred;
  npr_out[idx] = nv;
  int n = h * W_ + w;
  size_t nb = ((size_t)b * N_ + n) * D_;
  node[nb + c] = xv;
  node[nb + C_ + c] = nv;
}

// ---------------- Kernel 3: per-node L2 normalize --------------------------------
__global__ void k_norm(const float* __restrict__ node, float* __restrict__ nn) {
  int idx = blockIdx.x * blockDim.x + threadIdx.x;
  const int total = B_ * N_;
  if (idx >= total) return;
  const float* p = node + (size_t)idx * D_;
  float s = 0.f;
  for (int c = 0; c < D_; ++c) s += p[c] * p[c];
  float nrm = fmaxf(sqrtf(s), 1e-12f);
  float inv = 1.f / nrm;
  float* q = nn + (size_t)idx * D_;
  for (int c = 0; c < D_; ++c) q[c] = p[c] * inv;
}

// ---------------- Kernel 4: one fused walk step: T = rownorm(thresh(sim)) @ V ----
// One wave per 16-row tile. sim recomputed on the fly with f32 WMMA; tile is
// thresholded/diag-zeroed in C/D layout, bounced through LDS to A-operand
// layout, then multiplied into a 16x128 accumulator. Row sums reduced by
// lane butterflies; divide (rowsum + 1e-8) at the end.
// A tile for the sim GEMM (invariant across the column loop) is preloaded
// into 64 VGPRs so the hot loop issues only the B-operand load stream.
__global__ __launch_bounds__(32) void k_walk(const float* __restrict__ NNrm,
                                             const float* __restrict__ V,
                                             float* __restrict__ T) {
  int tile = blockIdx.x;
  int b = tile / (N_ / 16);
  int n0 = (tile % (N_ / 16)) * 16;
  int lane = threadIdx.x;
  int hlf = lane >> 4;   // which half-wave: K-offset group
  int l16 = lane & 15;

  const float* nnb = NNrm + (size_t)b * N_ * D_;
  const float* vb  = V    + (size_t)b * N_ * D_;

  __shared__ float simLDS[16 * 18];  // 16x16 tile, padded stride 18 (even -> b64 ok)

  v8f zero = {};
  v8f acc[8];
  float rowsum[8];
#pragma unroll
  for (int i = 0; i < 8; ++i) { acc[i] = zero; rowsum[i] = 0.f; }

  // ---- preload A operand tile (16x4 f32 layout, 32 K-steps) into registers ----
  const float* arow = nnb + (size_t)(n0 + l16) * D_ + 2 * hlf;  // 8B-aligned
  v2f apre[32];
#pragma unroll
  for (int kq = 0; kq < 32; ++kq)
    apre[kq] = *(const v2f*)(arow + kq * 4);

  for (int m0 = 0; m0 < N_; m0 += 16) {
    // ---- sim tile: node_norm[n0:n0+16] . node_norm[m0:m0+16]^T, K = 128 ----
    v8f s = zero;
    const float* brow = nnb + (size_t)(m0 + l16) * D_ + 2 * hlf;  // 8B-aligned
#pragma unroll
    for (int kq = 0; kq < 32; ++kq) {
      v2f bb = *(const v2f*)(brow + kq * 4);
      s = __builtin_amdgcn_wmma_f32_16x16x4_f32(false, apre[kq], false, bb,
                                                (short)0, s, false, false);
    }
    // ---- elementwise: threshold, zero diagonal; accumulate row sums; to LDS ----
#pragma unroll
    for (int r = 0; r < 8; ++r) {
      float v = s[r];
      int M = r + 8 * hlf;             // row within tile (C/D layout)
      v = (v >= THRESH_) ? v : 0.f;
      if (n0 + M == m0 + l16) v = 0.f; // zero diagonal
      rowsum[r] += v;
      simLDS[M * 18 + l16] = v;
    }
    __syncthreads();
    // ---- acc += simTile(16x16) @ V[m0:m0+16, :] (8 chunks of 16 cols) ----
#pragma unroll
    for (int dc = 0; dc < 8; ++dc) {
      v8f cacc = acc[dc];
#pragma unroll
      for (int kk = 0; kk < 4; ++kk) {
        // index is even -> 8-byte aligned -> single ds_load_b64
        v2f a = *(const v2f*)&simLDS[l16 * 18 + kk * 4 + 2 * hlf];
        const float* pv = vb + (size_t)(m0 + kk * 4 + 2 * hlf) * D_ + dc * 16 + l16;
        v2f bb; bb.x = pv[0]; bb.y = pv[D_];
        cacc = __builtin_amdgcn_wmma_f32_16x16x4_f32(false, a, false, bb,
                                                     (short)0, cacc, false, false);
      }
      acc[dc] = cacc;
    }
    __syncthreads();
  }

  // ---- reduce row sums across the 16 lanes of each half-wave ----
#pragma unroll
  for (int r = 0; r < 8; ++r) {
    float v = rowsum[r];
    for (int m = 1; m < 16; m <<= 1) v += __shfl_xor(v, m, 32);
    rowsum[r] = v;
  }

  float* tb = T + (size_t)b * N_ * D_;
#pragma unroll
  for (int dc = 0; dc < 8; ++dc) {
#pragma unroll
    for (int r = 0; r < 8; ++r) {
      float val = acc[dc][r] / (rowsum[r] + 1e-8f);
      tb[(size_t)(n0 + r + 8 * hlf) * D_ + dc * 16 + l16] = val;
    }
  }
}

// ---------------- Kernel 5: agg = 0.343 t3 + 0.147 t2 + 0.21 t1 + 0.3 node; transpose out
__global__ void k_combine(const float* __restrict__ node, const float* __restrict__ t1,
                          const float* __restrict__ t2, const float* __restrict__ t3,
                          float* __restrict__ gf) {
  int idx = blockIdx.x * blockDim.x + threadIdx.x;
  const int total = B_ * N_ * D_;
  if (idx >= total) return;
  int d = idx % D_;
  int n = (idx / D_) % N_;
  int b = idx / (D_ * N_);
  float v = 0.3f * node[idx] + 0.21f * t1[idx] + 0.147f * t2[idx] + 0.343f * t3[idx];
  gf[((size_t)b * D_ + d) * (size_t)N_ + n] = v;
}

extern "C" void kernel_launch(void* const* d_in, const int* in_sizes, int n_in,
                              void* d_out, int out_size, void* d_ws, size_t ws_size,
                              hipStream_t stream) {
  (void)in_sizes; (void)n_in; (void)out_size; (void)ws_size;
  const float* x = (const float*)d_in[0];
  float* out = (float*)d_out;
  float* npr = out;                                       // [B, C, H, W]
  float* gf  = out + (size_t)B_ * C_ * H_ * W_;           // [B, 2C, H, W]

  float* ws   = (float*)d_ws;
  float* down = ws;                                       // B*C*24*24
  float* node = down + (size_t)B_ * C_ * HD_ * WD_;       // B*N*128
  float* nn   = node + (size_t)B_ * N_ * D_;              // B*N*128
  float* t1   = nn   + (size_t)B_ * N_ * D_;
  float* t2   = t1   + (size_t)B_ * N_ * D_;
  float* t3   = t2   + (size_t)B_ * N_ * D_;

  {
    int tot = B_ * C_ * HD_ * WD_;
    k_down<<<(tot + 255) / 256, 256, 0, stream>>>(x, down);
  }
  {
    int tot = B_ * C_ * H_ * W_;
    k_npr_node<<<(tot + 255) / 256, 256, 0, stream>>>(x, down, npr, node);
  }
  {
    int tot = B_ * N_;
    k_norm<<<(tot + 255) / 256, 256, 0, stream>>>(node, nn);
  }
  dim3 wgrid(B_ * (N_ / 16));
  k_walk<<<wgrid, 32, 0, stream>>>(nn, node, t1);  // t1 = A @ node
  k_walk<<<wgrid, 32, 0, stream>>>(nn, t1, t2);    // t2 = A @ t1
  k_walk<<<wgrid, 32, 0, stream>>>(nn, t2, t3);    // t3 = A @ t2
  {
    int tot = B_ * N_ * D_;
    k_combine<<<(tot + 255) / 256, 256, 0, stream>>>(node, t1, t2, t3, gf);
  }
}